// model_own_20658792694596
// MI455X (gfx1250) — compile-verified
//
#include <hip/hip_runtime.h>
#include <cstdint>
#include <cstddef>

// ---------------------------------------------------------------------------
// Problem constants (match reference setup_inputs)
// ---------------------------------------------------------------------------
constexpr int BATCH = 8;
constexpr int NPTS  = 4096;
constexpr int KNN   = 20;
constexpr float BN_EPS  = 1e-5f;
constexpr float NEG_SLP = 0.2f;

typedef float v2f __attribute__((ext_vector_type(2)));
typedef float v8f __attribute__((ext_vector_type(8)));

__device__ __forceinline__ float lrelu_f(float y) { return y >= 0.0f ? y : NEG_SLP * y; }

// ---------------------------------------------------------------------------
// Utility: zero a float region (stats accumulators)
// ---------------------------------------------------------------------------
__global__ void dg_zero_kernel(float* __restrict__ p, int n) {
    int i = blockIdx.x * blockDim.x + threadIdx.x;
    if (i < n) p[i] = 0.0f;
}

// ---------------------------------------------------------------------------
// Squared norms per point:  xx[b*N+n] = sum_c X[b][c][n]^2
// ---------------------------------------------------------------------------
template<int C>
__global__ void dg_xx_kernel(const float* __restrict__ X, float* __restrict__ xx) {
    int gid = blockIdx.x * blockDim.x + threadIdx.x;
    if (gid >= BATCH * NPTS) return;
    int n = gid & (NPTS - 1);
    int b = gid >> 12;              // NPTS == 4096
    const float* Xb = X + (size_t)b * C * NPTS;
    float s = 0.0f;
#pragma unroll
    for (int c = 0; c < C; ++c) { float v = Xb[c * NPTS + n]; s += v * v; }
    xx[gid] = s;
}

// ---------------------------------------------------------------------------
// Fused KNN: WMMA f32 16x16x4 Gram tiles + register top-20 selection.
//   score(i,j) = 2*dot(x_i,x_j) - ||x_j||^2   (== ref d shifted by -||x_i||^2,
//   which preserves ordering AND exact-tie structure of top_k).
// One wave per 16-row M tile; 4 waves / 128-thread block; grid (N/64, B).
// A operand pre-scaled by 2 so the accumulator is 2*dot directly.
// ---------------------------------------------------------------------------
template<int CPAD, int CREAL>
__global__ __launch_bounds__(128)
void dg_knn_wmma_kernel(const float* __restrict__ X, const float* __restrict__ xx,
                        int* __restrict__ idx_out) {
    static_assert(CPAD % 4 == 0, "K chunks of 4");
    constexpr int NCH = CPAD / 4;

    const int b    = blockIdx.y;
    const int wave = threadIdx.x >> 5;
    const int lane = threadIdx.x & 31;
    const int hi   = lane >> 4;      // 0 or 1
    const int lo   = lane & 15;
    const int m0   = blockIdx.x * 64 + wave * 16;

    const float* Xb  = X  + (size_t)b * CREAL * NPTS;
    const float* xxb = xx + (size_t)b * NPTS;

    __shared__ float tile[4][16][16];   // private per wave
    __shared__ float xcol[4][16];

    // A tile: rows m0..m0+15, all channels, scaled by 2.
    // ISA A 16x4 layout: lane (M = lane%16), vgpr0 = K=2*(lane/16), vgpr1 = K+1.
    v2f areg[NCH];
#pragma unroll
    for (int c = 0; c < NCH; ++c) {
        int ch = 4 * c + 2 * hi;
        areg[c].x = (ch     < CREAL) ? 2.0f * Xb[(ch    ) * NPTS + m0 + lo] : 0.0f;
        areg[c].y = (ch + 1 < CREAL) ? 2.0f * Xb[(ch + 1) * NPTS + m0 + lo] : 0.0f;
    }

    // Per-row (lanes 0..15) top-20, sorted descending. Strict '>' during
    // insertion reproduces jax.lax.top_k tie-breaking (lower index wins)
    // because columns are scanned in ascending j order.
    float bd[KNN]; int bi[KNN];
#pragma unroll
    for (int t = 0; t < KNN; ++t) { bd[t] = -3.402823e38f; bi[t] = 0; }

    for (int j0 = 0; j0 < NPTS; j0 += 16) {
        __builtin_prefetch(Xb + j0 + 16 + lo, 0, 0);   // global_prefetch_b8

        v8f acc = {0.f,0.f,0.f,0.f,0.f,0.f,0.f,0.f};
#pragma unroll
        for (int c = 0; c < NCH; ++c) {
            int ch = 4 * c + 2 * hi;
            v2f breg;
            breg.x = (ch     < CREAL) ? Xb[(ch    ) * NPTS + j0 + lo] : 0.0f;
            breg.y = (ch + 1 < CREAL) ? Xb[(ch + 1) * NPTS + j0 + lo] : 0.0f;
            // D = A x B + C, f32 16x16x4  (8-arg f32/f16 pattern)
            acc = __builtin_amdgcn_wmma_f32_16x16x4_f32(
                false, areg[c], false, breg, (short)0, acc, false, false);
        }
        // C/D layout: lane col = lane%16, vgpr v -> row v + 8*(lane/16)
#pragma unroll
        for (int v = 0; v < 8; ++v)
            tile[wave][v + 8 * hi][lo] = acc[v];
        if (hi == 0) xcol[wave][lo] = xxb[j0 + lo];
        __syncthreads();

        if (hi == 0) {
            // lane 'lo' owns row m0+lo
            for (int jj = 0; jj < 16; ++jj) {
                float cand = tile[wave][lo][jj] - xcol[wave][jj];
                if (cand > bd[KNN - 1]) {
                    float dv = cand; int iv = j0 + jj;
#pragma unroll
                    for (int p = 0; p < KNN; ++p) {
                        if (dv > bd[p]) {
                            float td = bd[p]; bd[p] = dv; dv = td;
                            int   ti = bi[p]; bi[p] = iv; iv = ti;
                        }
                    }
                }
            }
        }
        __syncthreads();   // protect LDS tile before next iteration's stores
    }

    if (hi == 0) {
        int row = m0 + lo;
        int* out = idx_out + ((size_t)b * NPTS + row) * KNN;
#pragma unroll
        for (int t = 0; t < KNN; ++t) out[t] = bi[t];
    }
}

// ---------------------------------------------------------------------------
// Graph-feature builder: channels = [A_nbr-A_ctr (CA), A_ctr (CA),
//                                    B_nbr-B_ctr (CB), B_ctr (CB)]
// ---------------------------------------------------------------------------
template<int CA, int CB>
__device__ __forceinline__ void dg_build_feat(float* __restrict__ f,
                                              const float* __restrict__ XAb, int jA,
                                              const float* __restrict__ XBb, int jB,
                                              int n) {
#pragma unroll
    for (int c = 0; c < CA; ++c) {
        float ctr = XAb[c * NPTS + n];
        f[c]      = XAb[c * NPTS + jA] - ctr;
        f[CA + c] = ctr;
    }
    if constexpr (CB > 0) {
#pragma unroll
        for (int c = 0; c < CB; ++c) {
            float ctr = XBb[c * NPTS + n];
            f[2 * CA + c]      = XBb[c * NPTS + jB] - ctr;
            f[2 * CA + CB + c] = ctr;
        }
    }
}

// ---------------------------------------------------------------------------
// Conv pass 1: per-output-channel sum / sumsq of y = W * feat over (B,N,K).
// Thread per gathered column (b,n,kk). W staged in LDS; LDS then global atomics.
// ---------------------------------------------------------------------------
template<int O, int CA, int CB>
__global__ __launch_bounds__(256)
void dg_conv_stats_kernel(const float* __restrict__ W,
                          const float* __restrict__ XA, const int* __restrict__ idxA,
                          const float* __restrict__ XB, const int* __restrict__ idxB,
                          float* __restrict__ gsum, float* __restrict__ gsq) {
    constexpr int C = 2 * (CA + CB);
    __shared__ float s_w[O * C];
    __shared__ float s_sum[O], s_sq[O];
    for (int i = threadIdx.x; i < O * C; i += blockDim.x) s_w[i] = W[i];
    for (int i = threadIdx.x; i < O;     i += blockDim.x) { s_sum[i] = 0.f; s_sq[i] = 0.f; }
    __syncthreads();

    const int total = BATCH * NPTS * KNN;
    int col = blockIdx.x * blockDim.x + threadIdx.x;
    if (col < total) {
        int kk = col % KNN;
        int bn = col / KNN;
        int n  = bn & (NPTS - 1);
        int b  = bn >> 12;
        const float* XAb = XA + (size_t)b * CA * NPTS;
        const float* XBb = (CB > 0) ? (XB + (size_t)b * CB * NPTS) : nullptr;
        int jA = idxA[(size_t)bn * KNN + kk];
        int jB = (CB > 0) ? idxB[(size_t)bn * KNN + kk] : 0;

        float f[C];
        dg_build_feat<CA, CB>(f, XAb, jA, XBb, jB, n);

        for (int o = 0; o < O; ++o) {
            float y = 0.0f;
#pragma unroll
            for (int c = 0; c < C; ++c) y += s_w[o * C + c] * f[c];
            atomicAdd(&s_sum[o], y);
            atomicAdd(&s_sq[o],  y * y);
        }
    }
    __syncthreads();
    for (int i = threadIdx.x; i < O; i += blockDim.x) {
        atomicAdd(&gsum[i], s_sum[i]);
        atomicAdd(&gsq[i],  s_sq[i]);
    }
}

// ---------------------------------------------------------------------------
// BN finalize: a = g * rsqrt(var+eps), c = b - mu*a
// ---------------------------------------------------------------------------
__global__ void dg_bn_finalize_kernel(const float* __restrict__ gsum, const float* __restrict__ gsq,
                                      const float* __restrict__ gamma, const float* __restrict__ beta,
                                      float* __restrict__ a, float* __restrict__ c,
                                      int O, float invcnt) {
    int o = blockIdx.x * blockDim.x + threadIdx.x;
    if (o >= O) return;
    float mu  = gsum[o] * invcnt;
    float var = gsq[o] * invcnt - mu * mu;
    float s   = gamma[o] * rsqrtf(var + BN_EPS);
    a[o] = s;
    c[o] = beta[o] - mu * s;
}

// ---------------------------------------------------------------------------
// Conv pass 2: recompute y, apply BN + lrelu, max over kk.
// Thread per (b, n, o-group of 16). Output layout [b][o][n].
// ---------------------------------------------------------------------------
template<int O, int CA, int CB>
__global__ __launch_bounds__(128)
void dg_conv_apply_kernel(const float* __restrict__ W,
                          const float* __restrict__ XA, const int* __restrict__ idxA,
                          const float* __restrict__ XB, const int* __restrict__ idxB,
                          const float* __restrict__ bn_a, const float* __restrict__ bn_c,
                          float* __restrict__ out) {
    constexpr int C  = 2 * (CA + CB);
    constexpr int OG = 16;
    static_assert(O % OG == 0, "o-groups of 16");
    constexpr int NG = O / OG;

    __shared__ float s_w[O * C];
    __shared__ float s_ba[O], s_bc[O];
    for (int i = threadIdx.x; i < O * C; i += blockDim.x) s_w[i] = W[i];
    for (int i = threadIdx.x; i < O;     i += blockDim.x) { s_ba[i] = bn_a[i]; s_bc[i] = bn_c[i]; }
    __syncthreads();

    const int total = BATCH * NPTS * NG;
    int gid = blockIdx.x * blockDim.x + threadIdx.x;
    if (gid >= total) return;
    int og = gid % NG;
    int bn = gid / NG;
    int n  = bn & (NPTS - 1);
    int b  = bn >> 12;
    int obase = og * OG;

    const float* XAb = XA + (size_t)b * CA * NPTS;
    const float* XBb = (CB > 0) ? (XB + (size_t)b * CB * NPTS) : nullptr;

    float mx[OG];
#pragma unroll
    for (int o = 0; o < OG; ++o) mx[o] = -3.402823e38f;

    for (int kk = 0; kk < KNN; ++kk) {
        int jA = idxA[(size_t)bn * KNN + kk];
        int jB = (CB > 0) ? idxB[(size_t)bn * KNN + kk] : 0;
        float f[C];
        dg_build_feat<CA, CB>(f, XAb, jA, XBb, jB, n);
#pragma unroll
        for (int o = 0; o < OG; ++o) {
            float y = 0.0f;
#pragma unroll
            for (int c = 0; c < C; ++c) y += s_w[(obase + o) * C + c] * f[c];
            y = y * s_ba[obase + o] + s_bc[obase + o];
            y = lrelu_f(y);
            mx[o] = fmaxf(mx[o], y);
        }
    }
#pragma unroll
    for (int o = 0; o < OG; ++o)
        out[((size_t)b * O + obase + o) * NPTS + n] = mx[o];
}

// ---------------------------------------------------------------------------
// h = concat(xd1, xd2, ch).reshape(1,-1,128).mean(axis=1):
//   hmean[j] = mean over (b, c<128, m<32) of H[b][c][m*128+j]
// Block j (128 blocks), 256-thread tree reduction.
// ---------------------------------------------------------------------------
__global__ __launch_bounds__(256)
void dg_hmean_kernel(const float* __restrict__ xd1, const float* __restrict__ xd2,
                     float* __restrict__ hmean) {
    const int j = blockIdx.x;          // 0..127
    float s = 0.0f;
    // items: b(8) x c'(64) x m(32); add both xd1 and xd2 at same offset
    for (int it = threadIdx.x; it < 8 * 64 * 32; it += 256) {
        int m  = it & 31;
        int bc = it >> 5;              // b*64 + c'
        int n  = m * 128 + j;
        size_t off = (size_t)bc * NPTS + n;
        s += xd1[off] + xd2[off];
    }
    __shared__ float red[256];
    red[threadIdx.x] = s;
    __syncthreads();
    for (int st = 128; st > 0; st >>= 1) {
        if (threadIdx.x < st) red[threadIdx.x] += red[threadIdx.x + st];
        __syncthreads();
    }
    if (threadIdx.x == 0) hmean[j] = red[0] * (1.0f / 32768.0f);
}

// ---------------------------------------------------------------------------
// Dense head: 128 -> 128 -> 64 -> 11, lrelu after each (incl. last).
// ---------------------------------------------------------------------------
__global__ __launch_bounds__(128)
void dg_dense_kernel(const float* __restrict__ hmean,
                     const float* __restrict__ W2, const float* __restrict__ b2,
                     const float* __restrict__ W3, const float* __restrict__ b3,
                     const float* __restrict__ W4, const float* __restrict__ b4,
                     float* __restrict__ out) {
    __shared__ float h0[128], h1[128], h2[64];
    int t = threadIdx.x;
    h0[t] = hmean[t];
    __syncthreads();
    {
        float acc = b2[t];
        for (int c = 0; c < 128; ++c) acc += W2[t * 128 + c] * h0[c];
        h1[t] = lrelu_f(acc);
    }
    __syncthreads();
    if (t < 64) {
        float acc = b3[t];
        for (int c = 0; c < 128; ++c) acc += W3[t * 128 + c] * h1[c];
        h2[t] = lrelu_f(acc);
    }
    __syncthreads();
    if (t < 11) {
        float acc = b4[t];
        for (int c = 0; c < 64; ++c) acc += W4[t * 64 + c] * h2[c];
        out[t] = lrelu_f(acc);
    }
}

// ---------------------------------------------------------------------------
// Launch
// ---------------------------------------------------------------------------
extern "C" void kernel_launch(void* const* d_in, const int* in_sizes, int n_in,
                              void* d_out, int out_size, void* d_ws, size_t ws_size,
                              hipStream_t stream) {
    (void)in_sizes; (void)n_in; (void)out_size; (void)ws_size;

    const float* x   = (const float*)d_in[0];
    const float* w1  = (const float*)d_in[1];
    const float* g1  = (const float*)d_in[2];
    const float* b1  = (const float*)d_in[3];
    const float* w2  = (const float*)d_in[4];
    const float* g2  = (const float*)d_in[5];
    const float* b2  = (const float*)d_in[6];
    const float* wd1 = (const float*)d_in[7];
    const float* gd1 = (const float*)d_in[8];
    const float* bd1 = (const float*)d_in[9];
    const float* wd2 = (const float*)d_in[10];
    const float* gd2 = (const float*)d_in[11];
    const float* bd2 = (const float*)d_in[12];
    const float* W2d = (const float*)d_in[13];
    const float* b2d = (const float*)d_in[14];
    const float* W3d = (const float*)d_in[15];
    const float* b3d = (const float*)d_in[16];
    const float* W4d = (const float*)d_in[17];
    const float* b4d = (const float*)d_in[18];
    float* out = (float*)d_out;

    // ---- carve workspace ----
    char* ws = (char*)d_ws;
    size_t off = 0;
    auto carve = [&](size_t bytes) -> char* {
        char* p = ws + off;
        off = (off + bytes + 255) & ~(size_t)255;
        return p;
    };
    const size_t idx_elems = (size_t)BATCH * NPTS * KNN;
    int*   idx0  = (int*)  carve(idx_elems * 4);
    int*   idx1  = (int*)  carve(idx_elems * 4);
    int*   idx2  = (int*)  carve(idx_elems * 4);
    float* x1max = (float*)carve((size_t)BATCH * 32 * NPTS * 4);
    float* x2max = (float*)carve((size_t)BATCH * 64 * NPTS * 4);
    float* xd1   = (float*)carve((size_t)BATCH * 64 * NPTS * 4);
    float* xd2   = (float*)carve((size_t)BATCH * 64 * NPTS * 4);
    float* xx0   = (float*)carve((size_t)BATCH * NPTS * 4);
    float* xx1   = (float*)carve((size_t)BATCH * NPTS * 4);
    float* xx2   = (float*)carve((size_t)BATCH * NPTS * 4);
    float* stats = (float*)carve(1024 * 4);   // 4 convs x (sum[64] | sq[64])
    float* bnac  = (float*)carve(512 * 4);    // 4 convs x (a[64] | c[64])
    float* hmean = (float*)carve(128 * 4);

    float* sum1 = stats + 0 * 128, * sq1 = sum1 + 64;
    float* sum2 = stats + 1 * 128, * sq2 = sum2 + 64;
    float* sumA = stats + 2 * 128, * sqA = sumA + 64;
    float* sumB = stats + 3 * 128, * sqB = sumB + 64;
    float* a1 = bnac + 0 * 128, * c1 = a1 + 64;
    float* a2 = bnac + 1 * 128, * c2 = a2 + 64;
    float* aA = bnac + 2 * 128, * cA = aA + 64;
    float* aB = bnac + 3 * 128, * cB = aB + 64;

    const float invcnt = 1.0f / (float)(BATCH * NPTS * KNN);
    const int bnThreads = BATCH * NPTS;              // 32768
    const int colTotal  = BATCH * NPTS * KNN;        // 655360
    const dim3 knnGrid(NPTS / 64, BATCH);

    // 0) zero stats accumulators (ws is poisoned, not zeroed)
    dg_zero_kernel<<<4, 256, 0, stream>>>(stats, 1024);

    // 1) KNN on coords (pad C 3->4 inside the WMMA K-chunk)
    dg_xx_kernel<3><<<bnThreads / 256, 256, 0, stream>>>(x, xx0);
    dg_knn_wmma_kernel<4, 3><<<knnGrid, 128, 0, stream>>>(x, xx0, idx0);

    // 2) conv1: gf(x) [6ch] -> 32, BN, lrelu, max_k -> x1max
    dg_conv_stats_kernel<32, 3, 0><<<colTotal / 256, 256, 0, stream>>>(
        w1, x, idx0, nullptr, nullptr, sum1, sq1);
    dg_bn_finalize_kernel<<<1, 32, 0, stream>>>(sum1, sq1, g1, b1, a1, c1, 32, invcnt);
    dg_conv_apply_kernel<32, 3, 0><<<(bnThreads * 2) / 128, 128, 0, stream>>>(
        w1, x, idx0, nullptr, nullptr, a1, c1, x1max);

    // 3) KNN on x1max (C=32, WMMA)
    dg_xx_kernel<32><<<bnThreads / 256, 256, 0, stream>>>(x1max, xx1);
    dg_knn_wmma_kernel<32, 32><<<knnGrid, 128, 0, stream>>>(x1max, xx1, idx1);

    // 4) conv2: concat[gf(x) (6), gf(x1max) (64)] -> 64 -> x2max
    dg_conv_stats_kernel<64, 3, 32><<<colTotal / 256, 256, 0, stream>>>(
        w2, x, idx0, x1max, idx1, sum2, sq2);
    dg_bn_finalize_kernel<<<1, 64, 0, stream>>>(sum2, sq2, g2, b2, a2, c2, 64, invcnt);
    dg_conv_apply_kernel<64, 3, 32><<<(bnThreads * 4) / 128, 128, 0, stream>>>(
        w2, x, idx0, x1max, idx1, a2, c2, x2max);

    // 5) KNN on x2max (C=64, WMMA)
    dg_xx_kernel<64><<<bnThreads / 256, 256, 0, stream>>>(x2max, xx2);
    dg_knn_wmma_kernel<64, 64><<<knnGrid, 128, 0, stream>>>(x2max, xx2, idx2);

    // 6) dconv1: gf(x1max) [64] -> 64 -> xd1   (reuses idx1 = knn(x1max))
    dg_conv_stats_kernel<64, 32, 0><<<colTotal / 256, 256, 0, stream>>>(
        wd1, x1max, idx1, nullptr, nullptr, sumA, sqA);
    dg_bn_finalize_kernel<<<1, 64, 0, stream>>>(sumA, sqA, gd1, bd1, aA, cA, 64, invcnt);
    dg_conv_apply_kernel<64, 32, 0><<<(bnThreads * 4) / 128, 128, 0, stream>>>(
        wd1, x1max, idx1, nullptr, nullptr, aA, cA, xd1);

    // 7) dconv2: gf(x2max) [128] -> 64 -> xd2
    dg_conv_stats_kernel<64, 64, 0><<<colTotal / 256, 256, 0, stream>>>(
        wd2, x2max, idx2, nullptr, nullptr, sumB, sqB);
    dg_bn_finalize_kernel<<<1, 64, 0, stream>>>(sumB, sqB, gd2, bd2, aB, cB, 64, invcnt);
    dg_conv_apply_kernel<64, 64, 0><<<(bnThreads * 4) / 128, 128, 0, stream>>>(
        wd2, x2max, idx2, nullptr, nullptr, aB, cB, xd2);

    // 8) chunked mean (exact reference reshape semantics) + dense head
    dg_hmean_kernel<<<128, 256, 0, stream>>>(xd1, xd2, hmean);
    dg_dense_kernel<<<1, 128, 0, stream>>>(hmean, W2d, b2d, W3d, b3d, W4d, b4d, out);
}